// MaceLayer_54082228191974
// MI455X (gfx1250) — compile-verified
//
#include <hip/hip_runtime.h>
#include <math.h>

#define NAT 10000
#define NEDGE 160000

typedef __attribute__((ext_vector_type(2))) float v2f;
typedef __attribute__((ext_vector_type(8))) float v8f;

static __device__ __forceinline__ v8f wmma4f32(v2f a, v2f b, v8f c) {
  // D = A(16x4) * B(4x16) + C, full fp32 (V_WMMA_F32_16X16X4_F32)
  return __builtin_amdgcn_wmma_f32_16x16x4_f32(false, a, false, b, (short)0, c,
                                               false, false);
}

// Hardware fp32 global atomic add (GLOBAL_ATOMIC_ADD_F32, no return):
// unsafeAtomicAdd avoids the CAS loop the generic atomicAdd(float*) can emit.
static __device__ __forceinline__ void atomAddF(float* p, float v) {
  unsafeAtomicAdd(p, v);
}

// ---------------------------------------------------------------------------
// Kernel 1: per-edge geometry + Bessel RBF * polynomial cutoff
// ---------------------------------------------------------------------------
__global__ __launch_bounds__(256) void edge_geom(
    const float* __restrict__ pos, const int* __restrict__ ei,
    const float* __restrict__ freqs, float* __restrict__ rij,
    float* __restrict__ uvec) {
  int e = blockIdx.x * 256 + threadIdx.x;
  if (e >= NEDGE) return;
  int i = ei[e], j = ei[NEDGE + e];
  float dx = pos[j * 3 + 0] - pos[i * 3 + 0];
  float dy = pos[j * 3 + 1] - pos[i * 3 + 1];
  float dz = pos[j * 3 + 2] - pos[i * 3 + 2];
  float d = sqrtf(dx * dx + dy * dy + dz * dz + 1e-12f);
  float inv = 1.0f / d;
  uvec[e * 3 + 0] = dx * inv;
  uvec[e * 3 + 1] = dy * inv;
  uvec[e * 3 + 2] = dz * inv;
  float xs = d * (1.0f / 5.5f);
  float cut = 0.0f;
  if (xs < 1.0f) {
    float x2 = xs * xs, x3 = x2 * xs, x6 = x3 * x3;
    float x7 = x6 * xs, x8 = x7 * xs;
    cut = 1.0f - 28.0f * x6 + 48.0f * x7 - 21.0f * x8;
  }
  float pref = 0.603022689155527f * inv * cut;  // sqrt(2/5.5)
#pragma unroll
  for (int k = 0; k < 8; ++k) rij[e * 8 + k] = pref * sinf(freqs[k] * d);
}

// ---------------------------------------------------------------------------
// Kernel 2: generic 32x32 channel mix  Y[n,o,s] (+)= sum_c W[o,c] X[n,c,s]
// One wave handles 16 columns (col = n*S+s), M=32 via two 16x16 WMMA tiles,
// K=32 via eight 16x16x4 f32 WMMA steps. accumulate!=0 seeds C from Y.
// ---------------------------------------------------------------------------
__global__ __launch_bounds__(128) void mix32_wmma(
    const float* __restrict__ W, const float* __restrict__ X,
    float* __restrict__ Y, int N, int S, int accumulate) {
  const int lane = threadIdx.x & 31;
  const int wave = threadIdx.x >> 5;
  const int total = N * S;
  const int col0 = (blockIdx.x * 4 + wave) * 16;
  if (col0 >= total) return;

  const int r = lane & 15;             // B/D column within tile; A row base
  const int kof = (lane >> 4) * 2;     // K sub-offset (0 for lanes<16, 2 else)
  const int mbase = (lane >> 4) * 8;   // D: M = mbase + vgpr

  int bcol = col0 + r;
  bool bvalid = bcol < total;
  int bc = bvalid ? bcol : 0;
  int bn = bc / S, bs = bc % S;
  const float* Xc = X + (size_t)bn * 32 * S + bs;
  size_t ybase = (size_t)bn * 32 * S + bs;
  const float* W0 = W + r * 32 + kof;
  const float* W1 = W + (r + 16) * 32 + kof;

  v8f d0, d1;
  if (accumulate) {
#pragma unroll
    for (int t = 0; t < 8; ++t) {
      int o0 = mbase + t;
      d0[t] = bvalid ? Y[ybase + (size_t)o0 * S] : 0.0f;
      d1[t] = bvalid ? Y[ybase + (size_t)(o0 + 16) * S] : 0.0f;
    }
  } else {
#pragma unroll
    for (int t = 0; t < 8; ++t) {
      d0[t] = 0.0f;
      d1[t] = 0.0f;
    }
  }

#pragma unroll
  for (int k0 = 0; k0 < 32; k0 += 4) {
    float bx = Xc[(size_t)(k0 + kof) * S];
    float by = Xc[(size_t)(k0 + kof + 1) * S];
    v2f b;
    b.x = bvalid ? bx : 0.0f;
    b.y = bvalid ? by : 0.0f;
    v2f a0, a1;
    a0.x = W0[k0];
    a0.y = W0[k0 + 1];
    a1.x = W1[k0];
    a1.y = W1[k0 + 1];
    d0 = wmma4f32(a0, b, d0);
    d1 = wmma4f32(a1, b, d1);
  }

  if (bvalid) {
#pragma unroll
    for (int t = 0; t < 8; ++t) {
      int o0 = mbase + t;
      Y[ybase + (size_t)o0 * S] = d0[t];
      Y[ybase + (size_t)(o0 + 16) * S] = d1[t];
    }
  }
}

// ---------------------------------------------------------------------------
// Kernel 3: combined weights Cw[l] = Bu_w[l] @ B_w[l]   (folds two mixings)
// Cw layout: [32x192][32x192][32x224] concatenated.
// ---------------------------------------------------------------------------
__global__ __launch_bounds__(256) void cw_kernel(
    const float* __restrict__ Bu, const float* __restrict__ Bw0,
    const float* __restrict__ Bw1, const float* __restrict__ Bw2,
    float* __restrict__ Cw) {
  const int n0 = 32 * 192, n1 = 32 * 192, n2 = 32 * 224;
  int idx = blockIdx.x * 256 + threadIdx.x;
  if (idx >= n0 + n1 + n2) return;
  int l, off, Wd;
  const float* Bw;
  if (idx < n0) {
    l = 0; off = idx; Wd = 192; Bw = Bw0;
  } else if (idx < n0 + n1) {
    l = 1; off = idx - n0; Wd = 192; Bw = Bw1;
  } else {
    l = 2; off = idx - n0 - n1; Wd = 224; Bw = Bw2;
  }
  int o = off / Wd, k = off % Wd;
  const float* bu = Bu + l * 1024 + o * 32;
  float s = 0.0f;
#pragma unroll
  for (int c = 0; c < 32; ++c) s += bu[c] * Bw[c * Wd + k];
  Cw[idx] = s;
}

// ---------------------------------------------------------------------------
// Kernel 4: edge messages + scatter-add into A_pre.  lane = channel.
// EDGE_COMBOS (lx,ly,lz) index map:
// 0:(0,0,0) 1:(0,1,1) 2:(0,2,2) 3:(1,0,1) 4:(1,1,0) 5:(1,1,2)
// 6:(1,2,1) 7:(2,0,2) 8:(2,1,1) 9:(2,2,0) 10:(2,2,2)
// ---------------------------------------------------------------------------
__global__ __launch_bounds__(256) void edge_msg(
    const int* __restrict__ ei, const float* __restrict__ rijw,
    const float* __restrict__ uw, const float* __restrict__ hm,
    const float* __restrict__ rbfw, const float* __restrict__ rbfb,
    float* __restrict__ Ap) {
  __shared__ float lw[11 * 256];  // rbf_w transposed -> [i][k][c]
  __shared__ float lb[11 * 32];
  for (int t = threadIdx.x; t < 11 * 256; t += 256) {
    int i = t >> 8, rem = t & 255, c = rem >> 3, k = rem & 7;
    lw[i * 256 + k * 32 + c] = rbfw[t];
  }
  for (int t = threadIdx.x; t < 11 * 32; t += 256) lb[t] = rbfb[t];
  __syncthreads();

  const int lane = threadIdx.x & 31;
  int wid = (blockIdx.x * 256 + threadIdx.x) >> 5;
  int nw = (gridDim.x * 256) >> 5;
  const float* hm0 = hm;
  const float* hm1 = hm + (size_t)NAT * 32;
  const float* hm2 = hm + (size_t)NAT * 128;
  float* A0 = Ap;
  float* A1 = Ap + (size_t)NAT * 32;
  float* A2 = Ap + (size_t)NAT * 128;

  for (int e = wid; e < NEDGE; e += nw) {
    // prefetch next edge's gather rows (global_prefetch_b8) to hide L2 latency
    int e2 = e + nw;
    if (e2 < NEDGE) {
      int jj2 = ei[NEDGE + e2];
      __builtin_prefetch(&hm0[(size_t)jj2 * 32 + lane], 0, 1);
      __builtin_prefetch(&hm1[(size_t)jj2 * 96 + lane * 3], 0, 1);
      __builtin_prefetch(&hm2[(size_t)jj2 * 288 + lane * 9], 0, 1);
    }
    int ii = ei[e], jj = ei[NEDGE + e];
    float uu[3];
    uu[0] = uw[e * 3 + 0];
    uu[1] = uw[e * 3 + 1];
    uu[2] = uw[e * 3 + 2];
    float rb[8];
#pragma unroll
    for (int k = 0; k < 8; ++k) rb[k] = rijw[e * 8 + k];
    float r[11];
#pragma unroll
    for (int i = 0; i < 11; ++i) {
      float acc = lb[i * 32 + lane];
#pragma unroll
      for (int k = 0; k < 8; ++k) acc += rb[k] * lw[i * 256 + k * 32 + lane];
      r[i] = acc;
    }
    float y0 = hm0[(size_t)jj * 32 + lane];
    float y1[3], y2[9];
#pragma unroll
    for (int t = 0; t < 3; ++t) y1[t] = hm1[(size_t)jj * 96 + lane * 3 + t];
#pragma unroll
    for (int t = 0; t < 9; ++t) y2[t] = hm2[(size_t)jj * 288 + lane * 9 + t];

    float s1 = uu[0] * y1[0] + uu[1] * y1[1] + uu[2] * y1[2];
    float v2[3];
#pragma unroll
    for (int p = 0; p < 3; ++p)
      v2[p] = y2[p * 3] * uu[0] + y2[p * 3 + 1] * uu[1] + y2[p * 3 + 2] * uu[2];
    float uv2 = uu[0] * v2[0] + uu[1] * v2[1] + uu[2] * v2[2];

    float m0 = r[0] * y0 + r[4] * s1 + r[9] * uv2;
    atomAddF(&A0[(size_t)ii * 32 + lane], m0);
#pragma unroll
    for (int p = 0; p < 3; ++p) {
      float m1 = r[1] * y1[p] + r[3] * uu[p] * y0 + r[6] * v2[p] +
                 r[8] * uu[p] * s1;
      atomAddF(&A1[(size_t)ii * 96 + lane * 3 + p], m1);
    }
#pragma unroll
    for (int i2 = 0; i2 < 3; ++i2) {
#pragma unroll
      for (int q = 0; q < 3; ++q) {
        float m2 = r[2] * y2[i2 * 3 + q] + r[5] * uu[i2] * y1[q] +
                   r[7] * uu[i2] * uu[q] * y0 + r[10] * uu[i2] * v2[q];
        atomAddF(&A2[(size_t)ii * 288 + lane * 9 + i2 * 3 + q], m2);
      }
    }
  }
}

// ---------------------------------------------------------------------------
// Part-value evaluator for fused B stage. A = node base in LDS:
// A0(c)=A[c], A1(c,i)=A[32+c*3+i], A2(c,i,j)=A[128+c*9+i*3+j]
// Part orders match B_parts concatenation order in the reference.
// ---------------------------------------------------------------------------
static __device__ __forceinline__ float pv(const float* __restrict__ A, int l,
                                           int part, int c, int i, int q) {
  const float* a1 = A + 32 + c * 3;
  const float* a2 = A + 128 + c * 9;
  if (l == 0) {
    switch (part) {
      case 0: return A[c];
      case 1: return a1[0] * a1[0] + a1[1] * a1[1] + a1[2] * a1[2];
      case 2: {
        float s = 0.f;
        for (int m = 0; m < 9; ++m) s += a2[m] * a2[m];
        return s;
      }
      case 3: {
        float s = 0.f;
        for (int p = 0; p < 3; ++p)
          for (int m = 0; m < 3; ++m) s += a1[p] * a1[m] * a2[p * 3 + m];
        return s;
      }
      case 4: {
        float s = 0.f;
        for (int p = 0; p < 3; ++p) {
          float t = 0.f;
          for (int m = 0; m < 3; ++m) t += a1[m] * a2[p * 3 + m];
          s += a1[p] * t;
        }
        return s;
      }
      default: {  // sum_ij t222(i,j) A2(i,j)
        float s = 0.f;
        for (int p = 0; p < 3; ++p)
          for (int j = 0; j < 3; ++j) {
            float t = 0.f;
            for (int m = 0; m < 3; ++m) t += a2[p * 3 + m] * a2[j * 3 + m];
            s += t * a2[p * 3 + j];
          }
        return s;
      }
    }
  } else if (l == 1) {
    switch (part) {
      case 0: {
        float s = 0.f;
        for (int m = 0; m < 3; ++m) s += a1[m] * a2[q * 3 + m];
        return s;
      }
      case 1:
      case 2: {
        float t0 = a1[0] * a1[0] + a1[1] * a1[1] + a1[2] * a1[2];
        return t0 * a1[q];
      }
      case 3: {  // sum_p t121(p) A2(q,p)
        float s = 0.f;
        for (int p = 0; p < 3; ++p) {
          float t = 0.f;
          for (int m = 0; m < 3; ++m) t += a1[m] * a2[p * 3 + m];
          s += t * a2[q * 3 + p];
        }
        return s;
      }
      case 4: {
        float t = 0.f;
        for (int m = 0; m < 9; ++m) t += a2[m] * a2[m];
        return t * a1[q];
      }
      default: {  // sum_j t222(q,j) A1(j)
        float s = 0.f;
        for (int j = 0; j < 3; ++j) {
          float t = 0.f;
          for (int m = 0; m < 3; ++m) t += a2[q * 3 + m] * a2[j * 3 + m];
          s += t * a1[j];
        }
        return s;
      }
    }
  } else {
    switch (part) {
      case 0: return a1[i] * a1[q];
      case 1: {  // t222(i,q)
        float s = 0.f;
        for (int m = 0; m < 3; ++m) s += a2[i * 3 + m] * a2[q * 3 + m];
        return s;
      }
      case 2: {
        float t0 = a1[0] * a1[0] + a1[1] * a1[1] + a1[2] * a1[2];
        return t0 * a2[i * 3 + q];
      }
      case 3: {  // A1(i) * sum_j A1(j) A2(q,j)
        float s = 0.f;
        for (int m = 0; m < 3; ++m) s += a1[m] * a2[q * 3 + m];
        return a1[i] * s;
      }
      case 4: {  // t121(i) * A1(q)
        float t = 0.f;
        for (int m = 0; m < 3; ++m) t += a1[m] * a2[i * 3 + m];
        return t * a1[q];
      }
      case 5: {
        float t = 0.f;
        for (int m = 0; m < 9; ++m) t += a2[m] * a2[m];
        return t * a2[i * 3 + q];
      }
      default: {  // sum_j t222(i,j) A2(q,j)
        float s = 0.f;
        for (int j = 0; j < 3; ++j) {
          float t = 0.f;
          for (int m = 0; m < 3; ++m) t += a2[i * 3 + m] * a2[j * 3 + m];
          s += t * a2[q * 3 + j];
        }
        return s;
      }
    }
  }
}

// ---------------------------------------------------------------------------
// Kernel 5: fused product-basis + combined B mixing.
// upB[l] = Cw[l] @ concat(parts);  parts computed in-lane from LDS A tile.
// 1 wave / block, 16 nodes / block.  Writes d_out (later Hu pass accumulates).
// ---------------------------------------------------------------------------
__global__ __launch_bounds__(32) void b_fused(const float* __restrict__ Am,
                                              const float* __restrict__ Cw,
                                              float* __restrict__ out) {
  __shared__ float Al[16 * 416];
  const int lane = threadIdx.x;
  const int node0 = blockIdx.x * 16;
  const float* Am0 = Am;
  const float* Am1 = Am + (size_t)NAT * 32;
  const float* Am2 = Am + (size_t)NAT * 128;
  for (int t = lane; t < 16 * 416; t += 32) {
    int nd = t / 416, rr = t % 416;
    int g = node0 + nd;
    float v = 0.0f;
    if (g < NAT) {
      if (rr < 32)
        v = Am0[(size_t)g * 32 + rr];
      else if (rr < 128)
        v = Am1[(size_t)g * 96 + (rr - 32)];
      else
        v = Am2[(size_t)g * 288 + (rr - 128)];
    }
    Al[t] = v;
  }
  __syncthreads();

  const int r = lane & 15;
  const int kof = (lane >> 4) * 2;
  const int mbase = (lane >> 4) * 8;
  const float* Anode = &Al[r * 416];
  const int g = node0 + r;

  const int Sv[3] = {1, 3, 9};
  const int Pv[3] = {6, 6, 7};
  const int Wv[3] = {192, 192, 224};
  const int Cwo[3] = {0, 32 * 192, 2 * 32 * 192};
  const size_t Oo[3] = {0, (size_t)NAT * 32, (size_t)NAT * 128};

#pragma unroll
  for (int l = 0; l < 3; ++l) {
    const int S = Sv[l], P = Pv[l], Wd = Wv[l];
    const float* Cwl = Cw + Cwo[l];
    float* Ol = out + Oo[l];
    for (int s = 0; s < S; ++s) {
      const int si = (l == 2) ? (s / 3) : 0;
      const int sq = (l == 2) ? (s % 3) : s;
      v8f d0, d1;
#pragma unroll
      for (int t = 0; t < 8; ++t) {
        d0[t] = 0.0f;
        d1[t] = 0.0f;
      }
      for (int part = 0; part < P; ++part) {
#pragma unroll
        for (int k0 = 0; k0 < 32; k0 += 4) {
          int c0 = k0 + kof;
          v2f b;
          b.x = pv(Anode, l, part, c0, si, sq);
          b.y = pv(Anode, l, part, c0 + 1, si, sq);
          const float* w0 = Cwl + r * Wd + part * 32 + c0;
          const float* w1 = Cwl + (r + 16) * Wd + part * 32 + c0;
          v2f a0, a1;
          a0.x = w0[0];
          a0.y = w0[1];
          a1.x = w1[0];
          a1.y = w1[1];
          d0 = wmma4f32(a0, b, d0);
          d1 = wmma4f32(a1, b, d1);
        }
      }
      if (g < NAT) {
        size_t base = (size_t)g * 32 * S + s;
#pragma unroll
        for (int t = 0; t < 8; ++t) {
          int o0 = mbase + t;
          Ol[base + (size_t)o0 * S] = d0[t];
          Ol[base + (size_t)(o0 + 16) * S] = d1[t];
        }
      }
    }
  }
}

// ---------------------------------------------------------------------------
extern "C" void kernel_launch(void* const* d_in, const int* in_sizes, int n_in,
                              void* d_out, int out_size, void* d_ws,
                              size_t ws_size, hipStream_t stream) {
  (void)in_sizes; (void)n_in; (void)out_size; (void)ws_size;
  const float* positions = (const float*)d_in[0];
  const int* edge_index = (const int*)d_in[1];
  const float* h0 = (const float*)d_in[2];
  const float* h1 = (const float*)d_in[3];
  const float* h2 = (const float*)d_in[4];
  const float* freqs = (const float*)d_in[5];
  const float* rbf_w = (const float*)d_in[6];
  const float* rbf_b = (const float*)d_in[7];
  const float* elm_w = (const float*)d_in[8];
  const float* A_w = (const float*)d_in[9];
  const float* B_w0 = (const float*)d_in[10];
  const float* B_w1 = (const float*)d_in[11];
  const float* B_w2 = (const float*)d_in[12];
  const float* Bu_w = (const float*)d_in[13];
  const float* Hu_w = (const float*)d_in[14];
  float* out = (float*)d_out;
  float* ws = (float*)d_ws;

  // workspace layout (floats)
  float* rij = ws;                           // E*8
  float* uvec = rij + (size_t)NEDGE * 8;     // E*3
  float* hm = uvec + (size_t)NEDGE * 3;      // NAT*416 (elm-mixed h)
  float* Apre = hm + (size_t)NAT * 416;      // NAT*416
  float* Amix = Apre + (size_t)NAT * 416;    // NAT*416
  float* Cw = Amix + (size_t)NAT * 416;      // 19456

  hipMemsetAsync(Apre, 0, (size_t)NAT * 416 * sizeof(float), stream);

  edge_geom<<<(NEDGE + 255) / 256, 256, 0, stream>>>(positions, edge_index,
                                                     freqs, rij, uvec);

  const int Sv[3] = {1, 3, 9};
  const float* hin[3] = {h0, h1, h2};
  const size_t hoff[3] = {0, (size_t)NAT * 32, (size_t)NAT * 128};

  for (int l = 0; l < 3; ++l) {  // h = elm_w mix
    int tiles = (NAT * Sv[l] + 15) / 16;
    mix32_wmma<<<(tiles + 3) / 4, 128, 0, stream>>>(
        elm_w + l * 1024, hin[l], hm + hoff[l], NAT, Sv[l], 0);
  }

  cw_kernel<<<(32 * (192 + 192 + 224) + 255) / 256, 256, 0, stream>>>(
      Bu_w, B_w0, B_w1, B_w2, Cw);

  edge_msg<<<2500, 256, 0, stream>>>(edge_index, rij, uvec, hm, rbf_w, rbf_b,
                                     Apre);

  for (int l = 0; l < 3; ++l) {  // A = A_w mix
    int tiles = (NAT * Sv[l] + 15) / 16;
    mix32_wmma<<<(tiles + 3) / 4, 128, 0, stream>>>(
        A_w + l * 1024, Apre + hoff[l], Amix + hoff[l], NAT, Sv[l], 0);
  }

  b_fused<<<(NAT + 15) / 16, 32, 0, stream>>>(Amix, Cw, out);

  const size_t ooff[3] = {0, (size_t)NAT * 32, (size_t)NAT * 128};
  for (int l = 0; l < 3; ++l) {  // out += Hu_w @ hin
    int tiles = (NAT * Sv[l] + 15) / 16;
    mix32_wmma<<<(tiles + 3) / 4, 128, 0, stream>>>(
        Hu_w + l * 1024, hin[l], out + ooff[l], NAT, Sv[l], 1);
  }
}